// MTPLayerModel_13838384627918
// MI455X (gfx1250) — compile-verified
//
#include <hip/hip_runtime.h>
#include <hip/hip_bf16.h>
#include <math.h>

// ---------------------------------------------------------------------------
// MTP head for MI455X (gfx1250, wave32, WMMA).
// Memory-bound workload (lm_head_w stream = 1.06 GB @ 23.3 TB/s), so f32
// WMMA (V_WMMA_F32_16X16X4_F32) is used: exact math on the matrix pipe, and
// the ~5 TFLOPS needed to stay on the HBM roofline is trivially available.
// GEMM is software-pipelined with double-buffered LDS so the weight stream
// (HBM) and the WMMA pipe overlap every K-step.
// ---------------------------------------------------------------------------

typedef __attribute__((ext_vector_type(2))) float v2f;
typedef __attribute__((ext_vector_type(8))) float v8f;

#define HIDDEN 2048
#define VOCAB  129280
#define SEQ    1024
#define RMS_EPS 1e-6f

// ---------------- block reduction helpers (blockDim.x == 256) --------------
__device__ inline float block_reduce_sum256(float v, float* red) {
    const int t = threadIdx.x;
    red[t] = v; __syncthreads();
    #pragma unroll
    for (int off = 128; off > 0; off >>= 1) {
        if (t < off) red[t] += red[t + off];
        __syncthreads();
    }
    float r = red[0]; __syncthreads();
    return r;
}

__device__ inline float block_reduce_max256(float v, float* red) {
    const int t = threadIdx.x;
    red[t] = v; __syncthreads();
    #pragma unroll
    for (int off = 128; off > 0; off >>= 1) {
        if (t < off) red[t] = fmaxf(red[t], red[t + off]);
        __syncthreads();
    }
    float r = red[0]; __syncthreads();
    return r;
}

// ---------------- K1: gather + rmsnorm(emb) + rmsnorm(hid) + concat --------
__global__ __launch_bounds__(256)
void mtp_gather_norm_concat(const float* __restrict__ hidden,
                            const int*   __restrict__ input_ids,
                            const float* __restrict__ embed_w,
                            const float* __restrict__ enorm_w,
                            const float* __restrict__ hnorm_w,
                            float* __restrict__ combined) {
    __shared__ float red[256];
    const int s = blockIdx.x;
    const int t = threadIdx.x;
    float x[8];

    // embedding row (gather)
    const float* erow = embed_w + (size_t)input_ids[s] * HIDDEN;
    float ssq = 0.f;
    #pragma unroll
    for (int i = 0; i < 8; ++i) { x[i] = erow[t + i * 256]; ssq += x[i] * x[i]; }
    ssq = block_reduce_sum256(ssq, red);
    float r = rsqrtf(ssq * (1.0f / HIDDEN) + RMS_EPS);
    float* out = combined + (size_t)s * (2 * HIDDEN);
    #pragma unroll
    for (int i = 0; i < 8; ++i) {
        const int j = t + i * 256;
        out[j] = enorm_w[j] * x[i] * r;
    }

    // hidden row
    const float* hrow = hidden + (size_t)s * HIDDEN;
    ssq = 0.f;
    #pragma unroll
    for (int i = 0; i < 8; ++i) { x[i] = hrow[t + i * 256]; ssq += x[i] * x[i]; }
    ssq = block_reduce_sum256(ssq, red);
    r = rsqrtf(ssq * (1.0f / HIDDEN) + RMS_EPS);
    #pragma unroll
    for (int i = 0; i < 8; ++i) {
        const int j = t + i * 256;
        out[HIDDEN + j] = hnorm_w[j] * x[i] * r;
    }
}

// ---------------- K2/K4: NT GEMM via V_WMMA_F32_16X16X4_F32 ----------------
// Out[M][N] = A[M][K] * W[N][K]^T.  Block tile 128x128, K-step 32.
// 8 waves/block: wave (wm,wn) in a 2x4 grid, each owns 64x32 (4x2 WMMA tiles).
// Double-buffered LDS (2 x [128][36] per operand, 72 KB total) + register
// staging: global loads for tile k+1 are issued before the 64 WMMAs of tile
// k, so HBM latency is hidden behind the matrix pipe (one barrier / K-step).
__global__ __launch_bounds__(256)
void mtp_wmma_gemm_nt(const float* __restrict__ A,
                      const float* __restrict__ W,
                      float* __restrict__ Out,
                      int K, int N) {
    __shared__ float As[2][128][36];
    __shared__ float Bs[2][128][36];

    const int t    = threadIdx.x;
    const int lane = t & 31;
    const int wave = t >> 5;
    const int wm   = wave >> 2;           // 0..1  -> 64-row band
    const int wn   = wave & 3;            // 0..3  -> 32-col band
    const int l16  = lane & 15;
    const int koff = (lane >> 4) << 1;    // lanes 0-15 -> K{0,1}; 16-31 -> K{2,3}
    const int hoff = (lane >> 4) << 3;    // C/D: upper lanes hold M+8

    const int m0 = blockIdx.y * 128;
    const int n0 = blockIdx.x * 128;

    // per-thread tile-copy coordinates (1024 float4 slots per tile)
    int lrow[4], lcol[4];
    const float* abase[4];
    const float* bbase[4];
    #pragma unroll
    for (int it = 0; it < 4; ++it) {
        const int slot = t + it * 256;
        lrow[it] = slot >> 3;
        lcol[it] = (slot & 7) << 2;
        abase[it] = A + (size_t)(m0 + lrow[it]) * K + lcol[it];
        bbase[it] = W + (size_t)(n0 + lrow[it]) * K + lcol[it];
    }

    v8f acc[4][2] = {};
    float4 ra[4], rb[4];

    // prologue: stage tile 0 into LDS buffer 0
    #pragma unroll
    for (int it = 0; it < 4; ++it) {
        ra[it] = *(const float4*)(abase[it]);
        rb[it] = *(const float4*)(bbase[it]);
    }
    #pragma unroll
    for (int it = 0; it < 4; ++it) {
        *(float4*)&As[0][lrow[it]][lcol[it]] = ra[it];
        *(float4*)&Bs[0][lrow[it]][lcol[it]] = rb[it];
    }

    int buf = 0;
    for (int k0 = 0; k0 < K; k0 += 32, buf ^= 1) {
        // single barrier per K-step: makes buf's stores visible AND proves
        // every wave finished reading buf^1 (last read in iteration k-1)
        __syncthreads();

        const bool has_next = (k0 + 32 < K);
        if (has_next) {
            // issue next tile's global loads now; their wait lands just
            // before the ds_store below, overlapping HBM with the WMMAs
            #pragma unroll
            for (int it = 0; it < 4; ++it) {
                ra[it] = *(const float4*)(abase[it] + k0 + 32);
                rb[it] = *(const float4*)(bbase[it] + k0 + 32);
            }
        }
        if (k0 + 64 < K) {   // prime L2 two tiles ahead (global_prefetch_b8)
            __builtin_prefetch(abase[0] + k0 + 64, 0, 1);
            __builtin_prefetch(bbase[0] + k0 + 64, 0, 1);
        }

        // compute tile k from buffer `buf`
        #pragma unroll
        for (int kk = 0; kk < 8; ++kk) {      // 8 x K=4 sub-steps
            v2f a[4], b[2];
            #pragma unroll
            for (int mt = 0; mt < 4; ++mt)
                a[mt] = *(const v2f*)&As[buf][wm * 64 + mt * 16 + l16][kk * 4 + koff];
            #pragma unroll
            for (int nt = 0; nt < 2; ++nt)
                b[nt] = *(const v2f*)&Bs[buf][wn * 32 + nt * 16 + l16][kk * 4 + koff];
            #pragma unroll
            for (int mt = 0; mt < 4; ++mt)
                #pragma unroll
                for (int nt = 0; nt < 2; ++nt)
                    acc[mt][nt] = __builtin_amdgcn_wmma_f32_16x16x4_f32(
                        /*neg_a=*/false, a[mt], /*neg_b=*/false, b[nt],
                        /*c_mod=*/(short)0, acc[mt][nt],
                        /*reuse_a=*/false, /*reuse_b=*/false);
        }

        // park staged registers into the other buffer
        if (has_next) {
            #pragma unroll
            for (int it = 0; it < 4; ++it) {
                *(float4*)&As[buf ^ 1][lrow[it]][lcol[it]] = ra[it];
                *(float4*)&Bs[buf ^ 1][lrow[it]][lcol[it]] = rb[it];
            }
        }
    }

    // epilogue: 16x16 f32 C/D layout -> VGPR g holds M = g (+8 upper lanes)
    #pragma unroll
    for (int mt = 0; mt < 4; ++mt)
        #pragma unroll
        for (int nt = 0; nt < 2; ++nt) {
            const int col = n0 + wn * 32 + nt * 16 + l16;
            #pragma unroll
            for (int g = 0; g < 8; ++g) {
                const int row = m0 + wm * 64 + mt * 16 + hoff + g;
                Out[(size_t)row * N + col] = acc[mt][nt][g];
            }
        }
}

// ---------------- K3: rmsnorm(ln_w) then rmsnorm(norm_w) -------------------
__global__ __launch_bounds__(256)
void mtp_double_rmsnorm(const float* __restrict__ proj,
                        const float* __restrict__ ln_w,
                        const float* __restrict__ norm_w,
                        float* __restrict__ out) {
    __shared__ float red[256];
    const int s = blockIdx.x;
    const int t = threadIdx.x;
    const float* row = proj + (size_t)s * HIDDEN;
    float y[8];
    float ssq = 0.f;
    #pragma unroll
    for (int i = 0; i < 8; ++i) { y[i] = row[t + i * 256]; ssq += y[i] * y[i]; }
    ssq = block_reduce_sum256(ssq, red);
    float r = rsqrtf(ssq * (1.0f / HIDDEN) + RMS_EPS);
    #pragma unroll
    for (int i = 0; i < 8; ++i) y[i] = ln_w[t + i * 256] * y[i] * r;

    ssq = 0.f;
    #pragma unroll
    for (int i = 0; i < 8; ++i) ssq += y[i] * y[i];
    ssq = block_reduce_sum256(ssq, red);
    r = rsqrtf(ssq * (1.0f / HIDDEN) + RMS_EPS);
    float* orow = out + (size_t)s * HIDDEN;
    #pragma unroll
    for (int i = 0; i < 8; ++i) {
        const int j = t + i * 256;
        orow[j] = norm_w[j] * y[i] * r;
    }
}

// ---------------- K5: per-row shifted-NLL (log-softmax) --------------------
__global__ __launch_bounds__(256)
void mtp_nll_rows(const float* __restrict__ logits,
                  const int*   __restrict__ labels,
                  const float* __restrict__ loss_mask,
                  float* __restrict__ acc) {          // acc[0]=sum nll*m, acc[1]=sum m
    __shared__ float red[256];
    const int s = blockIdx.x;                         // 0..1022 (shifted rows)
    const int t = threadIdx.x;
    const float* row = logits + (size_t)s * VOCAB;

    float mx = -INFINITY;
    for (int j = t; j < VOCAB; j += 256) mx = fmaxf(mx, row[j]);
    mx = block_reduce_max256(mx, red);

    float se = 0.f;
    for (int j = t; j < VOCAB; j += 256) se += __expf(row[j] - mx);
    se = block_reduce_sum256(se, red);

    if (t == 0) {
        const int lbl = labels[s + 1];
        const float nll = -(row[lbl] - mx - __logf(se));
        const float m = loss_mask[s + 1];
        atomicAdd(&acc[0], nll * m);
        atomicAdd(&acc[1], m);
    }
}

__global__ void mtp_init_acc(float* acc) {
    if (threadIdx.x < 2) acc[threadIdx.x] = 0.0f;
}

__global__ void mtp_finalize_loss(const float* __restrict__ acc,
                                  float* __restrict__ out_loss) {
    out_loss[0] = acc[0] / (acc[1] + 1e-8f);
}

// ---------------------------------------------------------------------------
extern "C" void kernel_launch(void* const* d_in, const int* in_sizes, int n_in,
                              void* d_out, int out_size, void* d_ws, size_t ws_size,
                              hipStream_t stream) {
    const float* hidden_states = (const float*)d_in[0];
    const int*   input_ids     = (const int*)  d_in[1];
    const int*   labels        = (const int*)  d_in[2];
    const float* loss_mask     = (const float*)d_in[3];
    const float* embed_w       = (const float*)d_in[4];
    const float* enorm_w       = (const float*)d_in[5];
    const float* hnorm_w       = (const float*)d_in[6];
    const float* eh_proj_w     = (const float*)d_in[7];   // [2048][4096]
    const float* ln_w          = (const float*)d_in[8];
    const float* norm_w        = (const float*)d_in[9];
    const float* lm_head_w     = (const float*)d_in[10];  // [129280][2048]

    float* logits = (float*)d_out;                        // SEQ*VOCAB, then loss

    // workspace layout (~33.5 MB): combined | projected | normed | acc[2]
    float* combined  = (float*)d_ws;                      // SEQ * 4096
    float* projected = combined  + (size_t)SEQ * 2 * HIDDEN;  // SEQ * 2048
    float* normed    = projected + (size_t)SEQ * HIDDEN;      // SEQ * 2048
    float* acc       = normed    + (size_t)SEQ * HIDDEN;      // 2 floats

    // 1) gather + norms + concat
    mtp_gather_norm_concat<<<SEQ, 256, 0, stream>>>(
        hidden_states, input_ids, embed_w, enorm_w, hnorm_w, combined);

    // 2) projected = combined(1024x4096) @ eh_proj_w(2048x4096)^T
    mtp_wmma_gemm_nt<<<dim3((2 * HIDDEN) / 128, SEQ / 128), 256, 0, stream>>>(
        combined, eh_proj_w, projected, /*K=*/2 * HIDDEN, /*N=*/HIDDEN);

    // 3) normed = rmsnorm(rmsnorm(projected, ln_w), norm_w)
    mtp_double_rmsnorm<<<SEQ, 256, 0, stream>>>(projected, ln_w, norm_w, normed);

    // 4) logits = normed(1024x2048) @ lm_head_w(129280x2048)^T  (to d_out)
    mtp_wmma_gemm_nt<<<dim3(VOCAB / 128, SEQ / 128), 256, 0, stream>>>(
        normed, lm_head_w, logits, /*K=*/HIDDEN, /*N=*/VOCAB);

    // 5) masked mean NLL over shifted rows
    mtp_init_acc<<<1, 32, 0, stream>>>(acc);
    mtp_nll_rows<<<SEQ - 1, 256, 0, stream>>>(logits, labels, loss_mask, acc);
    mtp_finalize_loss<<<1, 1, 0, stream>>>(acc, logits + (size_t)SEQ * VOCAB);
}